// SequenceSmoothLossCTC_PAMI_90254442758694
// MI455X (gfx1250) — compile-verified
//
#include <hip/hip_runtime.h>
#include <cstddef>

// Problem constants (match reference setup_inputs)
#define NB    64          // batch
#define NT    512         // time steps
#define NC    1000        // classes
#define NS    32          // max target length
#define CORN  4           // candidates per sample
#define LL    65          // 2*NS+1 extended states
#define TCH   16          // timesteps per async chunk
#define PITCH 72          // padded row: [0..64]=emissions, [65..71]=lse copies (pad)
#define NCHUNK (NT / TCH) // 32
#define CHUNKV (TCH * PITCH)       // 1152
#define AITER  (CHUNKV / 128)      // 9 async loads per thread per chunk (uniform!)
#define NEGV  (-1.0e30f)

static_assert(CHUNKV % 128 == 0 && AITER == 9, "uniform async issue count");

__device__ __forceinline__ float lae2(float x, float y) {
  float m = fmaxf(x, y);
  if (m < -1.0e29f) return NEGV;
  return m + logf(expf(x - m) + expf(y - m));
}
__device__ __forceinline__ float lae3(float x, float y, float z) {
  float m = fmaxf(fmaxf(x, y), z);
  if (m < -1.0e29f) return NEGV;
  return m + logf(expf(x - m) + expf(y - m) + expf(z - m));
}

// ---------------------------------------------------------------------------
// Kernel 1: per (b,t) row logsumexp over NC=1000 classes.
// One wave32 per row, float4 (b128) streaming loads, shfl_xor reductions.
// This is the bandwidth-dominant pass: 131 MB streamed once (~5.6us at 23.3TB/s).
// ---------------------------------------------------------------------------
__global__ __launch_bounds__(256) void lse_kernel(const float* __restrict__ preds,
                                                  float* __restrict__ lse) {
  const int lane = threadIdx.x & 31;
  const int row  = blockIdx.x * 8 + (threadIdx.x >> 5);   // 8 waves per block
  const float4* p4 = reinterpret_cast<const float4*>(preds + (size_t)row * NC);
  float4 v[8];
  float m = -3.4e38f;
#pragma unroll
  for (int j = 0; j < 8; ++j) {
    const int idx = lane + j * 32;          // 250 float4 per row
    if (idx < 250) {                        // j<7 folds to always-true
      v[j] = p4[idx];
      m = fmaxf(m, fmaxf(fmaxf(v[j].x, v[j].y), fmaxf(v[j].z, v[j].w)));
    } else {
      v[j] = make_float4(0.f, 0.f, 0.f, 0.f);
    }
  }
#pragma unroll
  for (int off = 16; off > 0; off >>= 1) m = fmaxf(m, __shfl_xor(m, off));
  float s = 0.f;
#pragma unroll
  for (int j = 0; j < 8; ++j) {
    const int idx = lane + j * 32;
    if (idx < 250)
      s += expf(v[j].x - m) + expf(v[j].y - m) + expf(v[j].z - m) + expf(v[j].w - m);
  }
#pragma unroll
  for (int off = 16; off > 0; off >>= 1) s += __shfl_xor(s, off);
  if (lane == 0) lse[row] = m + logf(s);
}

// ---------------------------------------------------------------------------
// Async stage one chunk of emissions into LDS with CDNA5 async-to-LDS loads.
// Exactly AITER unconditional issues per thread => per-wave ASYNCcnt increment
// per chunk is a compile-time constant (AITER), enabling s_wait_asynccnt <= N
// pipelining with two chunks in flight.
// Row layout (PITCH=72): slot s<65 -> preds[b, t0+tt, ext[s]]; s>=65 -> lse[b,t0+tt].
// ---------------------------------------------------------------------------
__device__ __forceinline__ void issue_chunk_async(int tid, int t0, float* dstbase,
                                                  const float* prow, const float* lrow,
                                                  const int* ext_sh) {
#pragma unroll
  for (int j = 0; j < AITER; ++j) {
    const int idx = tid + j * 128;
    const int tt = idx / PITCH;
    const int s  = idx - tt * PITCH;
    const float* gsrc = (s < LL) ? (prow + (size_t)(t0 + tt) * NC + ext_sh[s])
                                 : (lrow + (t0 + tt));
    // low 32 bits of a generic LDS pointer == workgroup-relative LDS byte offset
    unsigned ldst = (unsigned)(unsigned long long)(const void*)(dstbase + idx);
    asm volatile("global_load_async_to_lds_b32 %0, %1, off"
                 :: "v"(ldst), "v"(gsrc) : "memory");
  }
}

// ---------------------------------------------------------------------------
// Kernel 2: CTC forward. One block (128 threads = 4 waves) per sequence.
// Wave 0 holds all 65 alpha states in registers (lane i owns states 2i,2i+1;
// lane 31 also owns state 64); s-1/s-2 deps via one __shfl_up. All 4 waves
// issue async gathers; double-buffered with TWO chunks in flight:
//   wait asynccnt<=AITER drains the older chunk (async loads complete in order).
// ---------------------------------------------------------------------------
__global__ __launch_bounds__(128) void ctc_forward_kernel(
    const float* __restrict__ preds, const int* __restrict__ text,
    const int* __restrict__ length, const int* __restrict__ stext,
    const int* __restrict__ slength, const float* __restrict__ lse,
    float* __restrict__ loss_out /* NB + NB*CORN entries */) {
  __shared__ float ebuf[2 * CHUNKV];
  __shared__ int   ext_sh[LL + 1];
  __shared__ float afin[LL];

  const int g    = blockIdx.x;
  const int tid  = threadIdx.x;
  const int lane = tid & 31;
  const int wave = tid >> 5;

  int bsel, tl;
  const int* tgt;
  if (g < NB) {
    bsel = g; tgt = text + (size_t)g * NS; tl = length[g];
  } else {
    int j = g - NB; bsel = j / CORN; tgt = stext + (size_t)j * NS; tl = slength[j];
  }

  if (tid < LL + 1) ext_sh[tid] = (tid < LL && (tid & 1)) ? tgt[tid >> 1] : 0;
  __syncthreads();

  const float* prow = preds + (size_t)bsel * NT * NC;
  const float* lrow = lse + (size_t)bsel * NT;

  // per-lane skip flag: state 2i+1 may take s-2 iff tgt[i] != tgt[i-1]
  const int  mytgt  = tgt[lane];
  const int  ptgt   = __shfl_up(mytgt, 1);
  const bool myskip = (lane > 0) && (mytgt != ptgt);

  // prologue: two chunks in flight
  issue_chunk_async(tid, 0,   &ebuf[0],      prow, lrow, ext_sh);
  issue_chunk_async(tid, TCH, &ebuf[CHUNKV], prow, lrow, ext_sh);

  float a0 = NEGV, a1 = NEGV, a2 = NEGV;  // alpha[2*lane], alpha[2*lane+1], alpha[64]@lane31

  for (int k = 0; k < NCHUNK; ++k) {
    // Drain chunk k: with chunk k+1 still outstanding (AITER insts/wave),
    // waiting <=AITER completes exactly the older chunk. Last chunk: wait 0.
    if (k == NCHUNK - 1) asm volatile("s_wait_asynccnt 0" ::: "memory");
    else                 asm volatile("s_wait_asynccnt 9" ::: "memory");
    __syncthreads();                         // chunk k visible to all waves

    if (wave == 0) {
      const float* buf = &ebuf[(k & 1) * CHUNKV];
      for (int tt = 0; tt < TCH; ++tt) {
        const int t = k * TCH + tt;
        const float lse_t = buf[tt * PITCH + LL];            // broadcast slot
        const float e0 = buf[tt * PITCH + 2 * lane]     - lse_t;
        const float e1 = buf[tt * PITCH + 2 * lane + 1] - lse_t;
        if (t == 0) {
          a0 = (lane == 0) ? e0 : NEGV;
          a1 = (lane == 0) ? e1 : NEGV;
          a2 = NEGV;
        } else {
          float am1 = __shfl_up(a1, 1);                      // alpha[2*lane-1]
          if (lane == 0) am1 = NEGV;
          const float n0 = lae2(a0, am1) + e0;               // even state: no skip
          const float n1 = lae3(a1, a0, myskip ? am1 : NEGV) + e1;
          if (lane == 31) {                                  // state 64 (even)
            const float e2 = buf[tt * PITCH + 64] - lse_t;
            a2 = lae2(a2, a1) + e2;                          // uses old a1
          }
          a0 = n0; a1 = n1;
        }
      }
    }
    __syncthreads();                         // wave0 done reading buf[k&1]
    if (k + 2 < NCHUNK)                      // refill the buffer just freed
      issue_chunk_async(tid, (k + 2) * TCH, &ebuf[(k & 1) * CHUNKV],
                        prow, lrow, ext_sh);
  }

  if (wave == 0) {
    afin[2 * lane] = a0;
    afin[2 * lane + 1] = a1;
    if (lane == 31) afin[64] = a2;
  }
  __syncthreads();
  if (tid == 0) {
    const int idx = 2 * tl;  // final blank; idx-1 = final label
    loss_out[g] = -lae2(afin[idx], afin[idx - 1]);
  }
}

// ---------------------------------------------------------------------------
// Kernel 3: combine per-sequence losses into the scalar output.
// ---------------------------------------------------------------------------
__global__ __launch_bounds__(256) void final_kernel(const float* __restrict__ loss_all,
                                                    const int* __restrict__ length,
                                                    const int* __restrict__ slength,
                                                    float* __restrict__ out) {
  __shared__ float red[256];
  const int tid = threadIdx.x;
  float acc = 0.f;
  if (tid < NB) {
    const float lm = loss_all[tid];
    acc += (lm / (float)length[tid]) * (1.0f / (float)NB);   // master mean
  }
  {
    const int j = tid;                       // 0..NB*CORN-1 == 0..255
    const float lm   = loss_all[j / CORN];   // detached master loss of sample
    const float conf = expf(-lm);
    const float d    = 1.f - conf;
    const float rank = 0.01f + 0.99f * d * d;   // SMOOTH_TAIL + (1-ST)*(1-conf)^2
    const float ls   = loss_all[NB + j];
    const float den  = fmaxf((float)slength[j], 1.f);
    acc += 0.1f * rank * ls / den * (1.0f / (float)(NB * CORN));  // ALPHA * mean
  }
  red[tid] = acc;
  __syncthreads();
  for (int off = 128; off > 0; off >>= 1) {
    if (tid < off) red[tid] += red[tid + off];
    __syncthreads();
  }
  if (tid == 0) out[0] = red[0];
}

extern "C" void kernel_launch(void* const* d_in, const int* in_sizes, int n_in,
                              void* d_out, int out_size, void* d_ws, size_t ws_size,
                              hipStream_t stream) {
  const float* preds   = (const float*)d_in[0];  // [64,512,1000] f32
  const int*   text    = (const int*)d_in[1];    // [64,32]
  const int*   length  = (const int*)d_in[2];    // [64]
  const int*   stext   = (const int*)d_in[3];    // [256,32]
  const int*   slength = (const int*)d_in[4];    // [256]

  float* lse  = (float*)d_ws;                    // NB*NT floats
  float* loss = lse + (size_t)NB * NT;           // NB + NB*CORN floats

  lse_kernel<<<(NB * NT) / 8, 256, 0, stream>>>(preds, lse);
  ctc_forward_kernel<<<NB + NB * CORN, 128, 0, stream>>>(preds, text, length, stext,
                                                         slength, lse, loss);
  final_kernel<<<1, 256, 0, stream>>>(loss, length, slength, (float*)d_out);
}